// HingeLoss_50362786512953
// MI455X (gfx1250) — compile-verified
//
#include <hip/hip_runtime.h>

typedef __attribute__((ext_vector_type(2))) float v2f;
typedef __attribute__((ext_vector_type(8))) float v8f;

#define NUM_NODES 100000
#define NUM_EDGES 1200000
#define EMB_DIM   64
#define ENC_NEG_INF 0x007FFFFFu   // order-preserving encoding of -inf

// Order-preserving float <-> uint mapping so float-max == uint-max.
__device__ __forceinline__ unsigned encF(float f) {
    unsigned u = __float_as_uint(f);
    return (u & 0x80000000u) ? ~u : (u | 0x80000000u);
}
__device__ __forceinline__ float decF(unsigned u) {
    unsigned b = (u & 0x80000000u) ? (u & 0x7FFFFFFFu) : ~u;
    return __uint_as_float(b);
}

__global__ void hinge_init(float* p_sum, unsigned* p_cnt, unsigned* n_enc, float* out) {
    int i = blockIdx.x * blockDim.x + threadIdx.x;
    if (i < NUM_NODES) {
        p_sum[i] = 0.f;
        p_cnt[i] = 0u;
        n_enc[i] = ENC_NEG_INF;
    }
    if (i == 0) out[0] = 0.f;
}

// One wave32 computes 16 edge dot-products via v_wmma_f32_16x16x4_f32.
// A (16x4 slice): rows = src vectors of edges [base..base+15]
//   lanes 0-15: v0=K+0, v1=K+1 ; lanes 16-31: v0=K+2, v1=K+3   (ISA f32 A layout)
// B (4x16 slice): cols = dst vectors, mirrored K split across lane halves.
// Diagonal of the 16x16 accumulator = the 16 per-edge dots.
template <int MODE> // 0 = pos (scatter sum+count), 1 = neg (scatter max)
__global__ void edge_dot_scatter(const float* __restrict__ emb,
                                 const int* __restrict__ nodes,
                                 const int* __restrict__ edges, // [2][NUM_EDGES]
                                 float* __restrict__ p_sum,
                                 unsigned* __restrict__ p_cnt,
                                 unsigned* __restrict__ n_enc) {
    const int tid  = blockIdx.x * blockDim.x + threadIdx.x;
    const int wave = tid >> 5;
    const int lane = tid & 31;
    const int half = lane >> 4;   // 0: K+0/1, 1: K+2/3
    const int lh   = lane & 15;   // row (A) / col (B) index within tile
    const int e    = wave * 16 + lh;   // NUM_EDGES % 16 == 0: no tail, EXEC stays all-1s

    const int src = edges[e];              // segment id for the scatter
    const int dst = edges[NUM_EDGES + e];
    const float* aRow = emb + (size_t)nodes[src] * EMB_DIM;
    const float* bRow = emb + (size_t)nodes[dst] * EMB_DIM;
    const int ko = half * 2;

    v8f c = {};
#pragma unroll
    for (int k = 0; k < EMB_DIM; k += 4) {
        v2f a = *(const v2f*)(aRow + k + ko);   // global_load_b64, 8B aligned
        v2f b = *(const v2f*)(bRow + k + ko);
        c = __builtin_amdgcn_wmma_f32_16x16x4_f32(
                /*neg_a=*/false, a, /*neg_b=*/false, b,
                /*c_mod=*/(short)0, c, /*reuse_a=*/false, /*reuse_b=*/false);
    }

    // Extract diagonal: C(m,m) lives in VGPR (m&7), lane m (m<8) or lane m+16 (m>=8).
    float myDot = 0.f;
#pragma unroll
    for (int m = 0; m < 16; ++m) {
        float v = __shfl(c[m & 7], (m < 8) ? m : (m + 16), 32);
        if (lane == m) myDot = v;
    }

    if (lane < 16) {
        if (MODE == 0) {
            atomicAdd(&p_sum[src], myDot);
            atomicAdd(&p_cnt[src], 1u);
        } else {
            atomicMax(&n_enc[src], encF(myDot));
        }
    }
}

__global__ void hinge_finalize(const float* __restrict__ p_sum,
                               const unsigned* __restrict__ p_cnt,
                               const unsigned* __restrict__ n_enc,
                               float* __restrict__ out) {
    int i = blockIdx.x * blockDim.x + threadIdx.x;
    float loss = 0.f;
    if (i < NUM_NODES) {
        float pd = p_sum[i] / fmaxf((float)p_cnt[i], 1.0f);
        unsigned ne = n_enc[i];
        float nd = (ne == ENC_NEG_INF) ? 0.f : decF(ne);   // empty segment -> 0
        loss = fmaxf(nd - pd + 1.0f, 0.f);                 // DELTA = 1.0, clip(min=0)
    }
    // wave32 reduction, then one atomic per wave
#pragma unroll
    for (int off = 16; off; off >>= 1) loss += __shfl_xor(loss, off, 32);
    if ((threadIdx.x & 31) == 0) atomicAdd(out, loss * (1.0f / NUM_NODES));
}

extern "C" void kernel_launch(void* const* d_in, const int* in_sizes, int n_in,
                              void* d_out, int out_size, void* d_ws, size_t ws_size,
                              hipStream_t stream) {
    const float* emb   = (const float*)d_in[0];
    const int*   nodes = (const int*)d_in[1];
    const int*   pos   = (const int*)d_in[2];
    const int*   neg   = (const int*)d_in[3];
    float* out = (float*)d_out;

    float*    p_sum = (float*)d_ws;
    unsigned* p_cnt = (unsigned*)((char*)d_ws + (size_t)NUM_NODES * 4);
    unsigned* n_enc = (unsigned*)((char*)d_ws + (size_t)NUM_NODES * 8);

    hinge_init<<<(NUM_NODES + 255) / 256, 256, 0, stream>>>(p_sum, p_cnt, n_enc, out);

    const int waves  = NUM_EDGES / 16;      // 75000 waves, 16 edges each
    const int blocks = waves * 32 / 256;    // 9375 blocks of 8 waves
    edge_dot_scatter<0><<<blocks, 256, 0, stream>>>(emb, nodes, pos, p_sum, p_cnt, n_enc);
    edge_dot_scatter<1><<<blocks, 256, 0, stream>>>(emb, nodes, neg, p_sum, p_cnt, n_enc);

    hinge_finalize<<<(NUM_NODES + 255) / 256, 256, 0, stream>>>(p_sum, p_cnt, n_enc, out);
}